// GCNPE_mean_78881369358549
// MI455X (gfx1250) — compile-verified
//
#include <hip/hip_runtime.h>

typedef __attribute__((ext_vector_type(2))) float v2f;
typedef __attribute__((ext_vector_type(8))) float v8f;

#define N_NODES 100000
#define N_EDGES 1600000
#define NFEAT   256
#define NHID    128
#define NCLASS  64

// ---------------------------------------------------------------------------
// CSR row-pointer build: edge_dst is sorted ascending, so row_ptr[i] =
// lower_bound(edge_dst, i). One thread per i in [0, N] (row_ptr has N+1 ents).
// ---------------------------------------------------------------------------
__global__ void build_rowptr(const int* __restrict__ dst, int* __restrict__ rowptr,
                             int nEdges, int nNodes) {
    int i = blockIdx.x * blockDim.x + threadIdx.x;
    if (i > nNodes) return;
    int lo = 0, hi = nEdges;
    while (lo < hi) {
        int mid = (lo + hi) >> 1;
        if (dst[mid] < i) lo = mid + 1; else hi = mid;
    }
    rowptr[i] = lo;
}

// ---------------------------------------------------------------------------
// FP32 WMMA GEMM: C[N x NDIM] = A[N x KDIM] @ B[KDIM x NDIM], row-major.
// One wave (32 lanes) owns a 16-row strip of A and sweeps all NDIM/16 column
// tiles, so A is read exactly once per strip; B (<=128KB) stays cached.
// Uses V_WMMA_F32_16X16X4_F32 (exact fp32 — matches reference precision).
//
// Fragment layouts (ISA 7.12.2, wave32):
//   A 16x4 : lane{lo,hi}: v.x=A[row0+lo][k+2hi], v.y=A[row0+lo][k+2hi+1]
//   B 4x16 : lane{lo,hi}: v.x=B[k+2hi][n0+lo],  v.y=B[k+2hi+1][n0+lo]
//   C 16x16: VGPR j -> row=row0+j+8hi, col=n0+lo
// ---------------------------------------------------------------------------
template <int KDIM, int NDIM>
__global__ __launch_bounds__(32)
void gemm_wmma_f32(const float* __restrict__ A, const float* __restrict__ B,
                   float* __restrict__ C) {
    constexpr int NT = NDIM / 16;
    const int lane = threadIdx.x & 31;
    const int lo   = lane & 15;
    const int hi   = lane >> 4;
    const int row0 = blockIdx.x * 16;   // N_NODES is a multiple of 16

    v8f acc[NT] = {};   // zero-init accumulators

    const float* __restrict__ arow = A + (size_t)(row0 + lo) * KDIM;

#pragma unroll 4
    for (int k = 0; k < KDIM; k += 4) {
        v2f a;
        a.x = arow[k + 2 * hi];
        a.y = arow[k + 2 * hi + 1];
        const float* __restrict__ brow = B + (size_t)(k + 2 * hi) * NDIM + lo;
#pragma unroll
        for (int t = 0; t < NT; ++t) {
            v2f b;
            b.x = brow[t * 16];
            b.y = brow[t * 16 + NDIM];
            // 8 args: (neg_a, A, neg_b, B, c_mod, C, reuse_a, reuse_b)
            acc[t] = __builtin_amdgcn_wmma_f32_16x16x4_f32(
                false, a, false, b, (short)0, acc[t], false, false);
        }
    }

#pragma unroll
    for (int t = 0; t < NT; ++t) {
#pragma unroll
        for (int j = 0; j < 8; ++j) {
            C[(size_t)(row0 + j + 8 * hi) * NDIM + t * 16 + lo] = acc[t][j];
        }
    }
}

// ---------------------------------------------------------------------------
// SpMM via CSR (sorted dst -> contiguous edge range per row), fused bias and
// optional ReLU. One block per destination node, one thread per feature.
// src[e]/w[e] are uniform within the block (scalar-cache friendly); H reads
// are coalesced across the block and mostly hit the 192MB L2 (H is 51MB).
// ---------------------------------------------------------------------------
__global__ void spmm_bias_act(const float* __restrict__ H, const int* __restrict__ rowptr,
                              const int* __restrict__ src, const float* __restrict__ wgt,
                              const float* __restrict__ bias, float* __restrict__ out,
                              int D, int do_relu) {
    const int i = blockIdx.x;
    const int f = threadIdx.x;
    const int e0 = rowptr[i];
    const int e1 = rowptr[i + 1];
    float acc = 0.0f;
    for (int e = e0; e < e1; ++e) {
        acc = fmaf(wgt[e], H[(size_t)src[e] * D + f], acc);
    }
    acc += bias[f];
    if (do_relu) acc = fmaxf(acc, 0.0f);
    out[(size_t)i * D + f] = acc;
}

// ---------------------------------------------------------------------------
// log_softmax over 64 classes: one wave per row, 2 elements per lane,
// wave32 shuffle reductions (max, then sum of exp).
// ---------------------------------------------------------------------------
__global__ __launch_bounds__(32)
void logsoftmax64(const float* __restrict__ Z, float* __restrict__ out) {
    const int i = blockIdx.x;
    const int t = threadIdx.x;  // 0..31
    const float a = Z[(size_t)i * 64 + t];
    const float b = Z[(size_t)i * 64 + t + 32];
    float m = fmaxf(a, b);
#pragma unroll
    for (int o = 16; o > 0; o >>= 1) m = fmaxf(m, __shfl_xor(m, o, 32));
    float s = expf(a - m) + expf(b - m);
#pragma unroll
    for (int o = 16; o > 0; o >>= 1) s += __shfl_xor(s, o, 32);
    const float lse = m + logf(s);
    out[(size_t)i * 64 + t]      = a - lse;
    out[(size_t)i * 64 + t + 32] = b - lse;
}

// mean of the three branch outputs -> finaloutput
__global__ void mean3(const float* __restrict__ o, float* __restrict__ out, int n) {
    int i = blockIdx.x * blockDim.x + threadIdx.x;
    if (i < n) out[i] = (o[i] + o[i + n] + o[i + 2 * n]) * (1.0f / 3.0f);
}

// ---------------------------------------------------------------------------
// Launcher. Inputs (setup_inputs order):
//  0:x 1:d1 2:d2 3:edge_src 4:edge_dst 5:edge_weight 6:W1 7:b1 8:W2 9:b2
// Output: [o1 | o2 | o3 | mean], each N_NODES*NCLASS fp32.
// Workspace: row_ptr (0.4MB) + bufA (51.2MB) + bufB (51.2MB), ping-ponged:
//  GEMM1 feat->bufA, SpMM1 bufA->bufB(relu), GEMM2 bufB->bufA,
//  SpMM2 bufA->bufB, softmax bufB->d_out.
// ---------------------------------------------------------------------------
extern "C" void kernel_launch(void* const* d_in, const int* in_sizes, int n_in,
                              void* d_out, int out_size, void* d_ws, size_t ws_size,
                              hipStream_t stream) {
    const float* x   = (const float*)d_in[0];
    const float* dr1 = (const float*)d_in[1];
    const float* dr2 = (const float*)d_in[2];
    const int*   esrc = (const int*)d_in[3];
    const int*   edst = (const int*)d_in[4];
    const float* ew   = (const float*)d_in[5];
    const float* W1   = (const float*)d_in[6];
    const float* b1   = (const float*)d_in[7];
    const float* W2   = (const float*)d_in[8];
    const float* b2   = (const float*)d_in[9];
    float* out = (float*)d_out;

    char* ws = (char*)d_ws;
    int* rowptr = (int*)ws;
    size_t off = (((size_t)(N_NODES + 1) * sizeof(int)) + 255) & ~(size_t)255;
    float* bufA = (float*)(ws + off);
    float* bufB = bufA + (size_t)N_NODES * NHID;

    build_rowptr<<<(N_NODES + 1 + 255) / 256, 256, 0, stream>>>(edst, rowptr,
                                                                N_EDGES, N_NODES);

    const float* feats[3] = {x, dr1, dr2};
    for (int b = 0; b < 3; ++b) {
        gemm_wmma_f32<NFEAT, NHID><<<N_NODES / 16, 32, 0, stream>>>(feats[b], W1, bufA);
        spmm_bias_act<<<N_NODES, NHID, 0, stream>>>(bufA, rowptr, esrc, ew, b1,
                                                    bufB, NHID, 1);
        gemm_wmma_f32<NHID, NCLASS><<<N_NODES / 16, 32, 0, stream>>>(bufB, W2, bufA);
        spmm_bias_act<<<N_NODES, NCLASS, 0, stream>>>(bufA, rowptr, esrc, ew, b2,
                                                      bufB, NCLASS, 0);
        logsoftmax64<<<N_NODES, 32, 0, stream>>>(bufB,
                                                 out + (size_t)b * N_NODES * NCLASS);
    }

    const int n = N_NODES * NCLASS;
    mean3<<<(n + 255) / 256, 256, 0, stream>>>(out, out + (size_t)3 * n, n);
}